// Select_22763326669408
// MI455X (gfx1250) — compile-verified
//
#include <hip/hip_runtime.h>
#include <hip/hip_bf16.h>
#include <stdint.h>

#define N_ROWS 16384
#define D_DIM  4096
#define C_DIM  1024
#define N_HALF 8192

// ---------------------------------------------------------------------------
// Kernel 1: per-row loss.  One wave32 per row, 8 waves (256 threads) per block.
// loss[i] = | sum_c label[i,c] * (log(score1[i,c]) - log(score2[i,c])) |
// ---------------------------------------------------------------------------
__global__ __launch_bounds__(256) void loss_kernel(const float* __restrict__ score1,
                                                   const float* __restrict__ score2,
                                                   const float* __restrict__ label,
                                                   float* __restrict__ loss) {
    const int wave = threadIdx.x >> 5;   // 0..7
    const int lane = threadIdx.x & 31;   // wave32
    const int row  = blockIdx.x * 8 + wave;
    if (row >= N_ROWS) return;

    const float4* p1 = (const float4*)(score1 + (size_t)row * C_DIM);
    const float4* p2 = (const float4*)(score2 + (size_t)row * C_DIM);
    const float4* pl = (const float4*)(label  + (size_t)row * C_DIM);

    float acc = 0.0f;
#pragma unroll
    for (int k = 0; k < (C_DIM / 4) / 32; ++k) {   // 8 iterations
        const int c4 = k * 32 + lane;              // contiguous 512B per wave
        float4 a = p1[c4];
        float4 b = p2[c4];
        float4 w = pl[c4];
        acc += w.x * (logf(a.x) - logf(b.x));
        acc += w.y * (logf(a.y) - logf(b.y));
        acc += w.z * (logf(a.z) - logf(b.z));
        acc += w.w * (logf(a.w) - logf(b.w));
    }
    // wave32 butterfly reduction (never assume wave64 on gfx1250)
#pragma unroll
    for (int off = 16; off > 0; off >>= 1)
        acc += __shfl_xor(acc, off, 32);

    if (lane == 0) loss[row] = fabsf(acc);
}

// ---------------------------------------------------------------------------
// Kernel 2: exact stable rank of every row among ascending losses.
// rank(i) = #{ j : loss[j] < loss[i]  ||  (loss[j]==loss[i] && j < i) }
// Ranks are a permutation of 0..N-1, so idx[rank]=i (rank<N/2) writes each
// slot of idx exactly once.  Matches lax.top_k(-loss) ordering incl. ties.
// Grid: 64 blocks x 256 threads; whole loss table cached in 64KB LDS.
// ---------------------------------------------------------------------------
__global__ __launch_bounds__(256) void rank_kernel(const float* __restrict__ loss,
                                                   int* __restrict__ idx) {
    __shared__ __align__(16) float L[N_ROWS];     // 64 KB (WGP has 320 KB)
    const int t = threadIdx.x;

    const float4* Lg = (const float4*)loss;
    float4*       Ls = (float4*)L;
#pragma unroll
    for (int k = 0; k < (N_ROWS / 4) / 256; ++k)  // 16 float4 per thread
        Ls[t + k * 256] = Lg[t + k * 256];
    __syncthreads();

    const int   i  = blockIdx.x * 256 + t;
    const float mi = L[i];
    int rank = 0;
    for (int j4 = 0; j4 < N_ROWS / 4; ++j4) {     // broadcast LDS reads
        const float4 v = Ls[j4];
        const int j = j4 * 4;
        rank += (v.x < mi) || (v.x == mi && (j + 0) < i);
        rank += (v.y < mi) || (v.y == mi && (j + 1) < i);
        rank += (v.z < mi) || (v.z == mi && (j + 2) < i);
        rank += (v.w < mi) || (v.w == mi && (j + 3) < i);
    }
    if (rank < N_HALF) idx[rank] = i;
}

// ---------------------------------------------------------------------------
// CDNA5 async global<->LDS movers (gfx1250; ASYNCcnt-tracked).
// LDS byte address = low 32 bits of the generic pointer (ISA 10.2 aperture).
// ---------------------------------------------------------------------------
__device__ __forceinline__ void async_load_to_lds_b128(uint32_t lds_addr,
                                                       const void* gaddr) {
    asm volatile("global_load_async_to_lds_b128 %0, %1, off"
                 :: "v"(lds_addr), "v"(gaddr) : "memory");
}
__device__ __forceinline__ void async_store_from_lds_b128(void* gaddr,
                                                          uint32_t lds_addr) {
    asm volatile("global_store_async_from_lds_b128 %0, %1, off"
                 :: "v"(gaddr), "v"(lds_addr) : "memory");
}
__device__ __forceinline__ void wait_asynccnt0() {
    asm volatile("s_wait_asynccnt 0x0" ::: "memory");
}

// ---------------------------------------------------------------------------
// Kernel 3: gather.  One block (8 waves) per output row r:
//   out[r*D .. ]            = input2[idx[r]*D .. ]   (16 KB)
//   out[NH*D + r*C .. ]     = label [idx[r]*C .. ]   ( 4 KB)
// Fully pipelined async streaming: each lane owns 5 disjoint 16B LDS slots
// (one per chunk, 20KB/block).  Issue all 5 async loads, a single
// s_wait_asynccnt 0 (async loads complete in order; no LDS slot is reused so
// there is no WAR hazard), then issue all 5 async stores.  S_ENDPGM performs
// an implicit wait-idle, so no trailing wait is required.
// ---------------------------------------------------------------------------
__global__ __launch_bounds__(256) void gather_kernel(const float* __restrict__ input2,
                                                     const float* __restrict__ label,
                                                     const int* __restrict__ idx,
                                                     float* __restrict__ out) {
    __shared__ __align__(16) float buf[256 * 4 * 5];  // 20 KB staging (5 slots)
    const int r = blockIdx.x;
    const int t = threadIdx.x;
    const int s = idx[r];                             // uniform -> scalar load

    const uint32_t lds0 = (uint32_t)(uintptr_t)(&buf[0]) + (uint32_t)t * 16u;

    const float* src  = input2 + (size_t)s * D_DIM;
    float*       dst  = out    + (size_t)r * D_DIM;
    const float* srcL = label  + (size_t)s * C_DIM;
    float*       dstL = out + (size_t)N_HALF * D_DIM + (size_t)r * C_DIM;

    // ---- issue all async loads (5 in flight per wave) ----
#pragma unroll
    for (int k = 0; k < 4; ++k)
        async_load_to_lds_b128(lds0 + (uint32_t)k * 4096u, src + k * 1024 + t * 4);
    async_load_to_lds_b128(lds0 + 4u * 4096u, srcL + t * 4);

    wait_asynccnt0();   // all loads landed in LDS

    // ---- issue all async stores; implicit wait at S_ENDPGM completes them ----
#pragma unroll
    for (int k = 0; k < 4; ++k)
        async_store_from_lds_b128(dst + k * 1024 + t * 4, lds0 + (uint32_t)k * 4096u);
    async_store_from_lds_b128(dstL + t * 4, lds0 + 4u * 4096u);
}

// ---------------------------------------------------------------------------
// Host launcher.  Inputs (setup_inputs order): input1, input2, score1,
// score2, label — all float32.  d_out = [inputss (8192x4096) | labelss
// (8192x1024)] flat f32.  d_ws: loss[16384] f32 @0, idx[8192] i32 @64KB.
// ---------------------------------------------------------------------------
extern "C" void kernel_launch(void* const* d_in, const int* in_sizes, int n_in,
                              void* d_out, int out_size, void* d_ws, size_t ws_size,
                              hipStream_t stream) {
    (void)in_sizes; (void)n_in; (void)out_size; (void)ws_size;
    // const float* input1 = (const float*)d_in[0];   // unused by the op
    const float* input2 = (const float*)d_in[1];
    const float* score1 = (const float*)d_in[2];
    const float* score2 = (const float*)d_in[3];
    const float* label  = (const float*)d_in[4];

    float* loss = (float*)d_ws;
    int*   idx  = (int*)((char*)d_ws + (size_t)N_ROWS * sizeof(float));
    float* outp = (float*)d_out;

    loss_kernel<<<N_ROWS / 8, 256, 0, stream>>>(score1, score2, label, loss);
    rank_kernel<<<N_ROWS / 256, 256, 0, stream>>>(loss, idx);
    gather_kernel<<<N_HALF, 256, 0, stream>>>(input2, label, idx, outp);
}